// Proposal_layer_11450382811676
// MI455X (gfx1250) — compile-verified
//
#include <hip/hip_runtime.h>
#include <hip/hip_bf16.h>
#include <stdint.h>

// ---------------------------------------------------------------------------
// RPN proposal layer for MI455X (gfx1250, wave32).
//   K1 key build     : 589824 anchors -> u32 sortable score key only (boxes
//                      are NOT materialized: only 2% would ever be read)
//   K2 radix-select  : 3-level histogram select (11+11+10 bits) -> exact
//                      threshold T for the 12000th largest key
//   K3 compact       : keys >= T -> (~score,index) u64 pairs (cap 16384)
//   K4 sort+gather   : ONE workgroup, 128KB LDS bitonic sort of 16384 pairs
//                      (CDNA5 320KB/WGP) -> exact top_k order; re-decodes the
//                      12000 winning boxes from delta (L2-resident)
//   K5 nms           : single 1024-thread workgroup; 12000 boxes + areas in
//                      240KB LDS, filled via global_load_async_to_lds_b128
//                      (ASYNCcnt), greedy sequential NMS, zero-filled tail.
// Total global traffic ~26MB; runtime dominated by the inherently serial NMS.
// ---------------------------------------------------------------------------

#define N_ANCHORS 589824      // 256*256*9
#define CAP       16384u      // compaction capacity (>= 12000 + ties)
#define PRE_NMS   12000
#define POST_NMS  2000
#define IMG_W     4096.0f
#define IMG_H     4096.0f
#define MIN_SZ    16.0f
#define NMS_TH    0.7f

typedef unsigned long long u64;

// 9 base anchors (x0,y0,x1,y1) from ratios{0.5,1,2} x scales{8,16,32}
__constant__ float AX0[9] = {-84.f,-176.f,-360.f,-56.f,-120.f,-248.f,-36.f,-80.f,-168.f};
__constant__ float AY0[9] = {-40.f,-88.f,-184.f,-56.f,-120.f,-248.f,-80.f,-168.f,-344.f};
__constant__ float AX1[9] = { 99.f, 191.f, 375.f, 71.f, 135.f, 263.f, 51.f, 95.f, 183.f};
__constant__ float AY1[9] = { 55.f, 103.f, 199.f, 71.f, 135.f, 263.f, 95.f, 183.f, 359.f};

// state[] indices
#define ST_SEL1 0
#define ST_CA1  1
#define ST_SEL2 2
#define ST_CA2  3
#define ST_T    4
#define ST_CNT  5

// ---- shared box decode: anchor n + delta -> clipped proposal ---------------
__device__ __forceinline__ float4 decode_box(const float* __restrict__ delta, unsigned n) {
    unsigned a    = n % 9u;
    unsigned cell = n / 9u;
    float sx = (float)(cell % 256u) * 16.0f;
    float sy = (float)(cell / 256u) * 16.0f;

    float x0 = AX0[a] + sx, y0 = AY0[a] + sy;
    float x1 = AX1[a] + sx, y1 = AY1[a] + sy;
    float w  = x1 - x0 + 1.0f;
    float h  = y1 - y0 + 1.0f;
    float cx = x0 + 0.5f * w;
    float cy = y0 + 0.5f * h;

    float d0 = delta[4*n+0], d1 = delta[4*n+1], d2 = delta[4*n+2], d3 = delta[4*n+3];
    float pcx = d0 * w + cx;
    float pcy = d1 * h + cy;
    float pw  = __expf(d2) * w;
    float ph  = __expf(d3) * h;

    float px0 = fminf(fmaxf(pcx - 0.5f * pw, 0.0f), IMG_W);
    float py0 = fminf(fmaxf(pcy - 0.5f * ph, 0.0f), IMG_H);
    float px1 = fminf(fmaxf(pcx + 0.5f * pw, 0.0f), IMG_W);
    float py1 = fminf(fmaxf(pcy + 0.5f * ph, 0.0f), IMG_H);
    return make_float4(px0, py0, px1, py1);
}

// ---------------- K1: sortable key build (no box store) ---------------------
__global__ void key_kernel(const float* __restrict__ delta,
                           const float* __restrict__ score,
                           unsigned* __restrict__ sortable) {
    unsigned n = blockIdx.x * blockDim.x + threadIdx.x;
    if (n >= N_ANCHORS) return;

    float4 p = decode_box(delta, n);
    bool valid = ((p.z - p.x) >= MIN_SZ) && ((p.w - p.y) >= MIN_SZ);

    float s = score[2*n + 1];
    unsigned b = __float_as_uint(s);
    unsigned key = (b & 0x80000000u) ? ~b : (b | 0x80000000u); // order-preserving
    if (!valid) key = 0u;                                      // == -inf marker
    sortable[n] = key;
}

// ---------------- K2a/b/c: histogram passes (LDS-privatized) ----------------
__global__ __launch_bounds__(256) void hist1_kernel(const unsigned* __restrict__ sortable,
                                                    unsigned* __restrict__ hist) {
    __shared__ unsigned lh[2048];
    for (int t = threadIdx.x; t < 2048; t += 256) lh[t] = 0u;
    __syncthreads();
    for (unsigned n = blockIdx.x * blockDim.x + threadIdx.x; n < N_ANCHORS;
         n += gridDim.x * blockDim.x)
        atomicAdd(&lh[sortable[n] >> 21], 1u);
    __syncthreads();
    for (int t = threadIdx.x; t < 2048; t += 256)
        if (lh[t]) atomicAdd(&hist[t], lh[t]);
}

__global__ __launch_bounds__(256) void hist2_kernel(const unsigned* __restrict__ sortable,
                                                    const unsigned* __restrict__ state,
                                                    unsigned* __restrict__ hist) {
    __shared__ unsigned lh[2048];
    for (int t = threadIdx.x; t < 2048; t += 256) lh[t] = 0u;
    __syncthreads();
    const unsigned sel1 = state[ST_SEL1];
    for (unsigned n = blockIdx.x * blockDim.x + threadIdx.x; n < N_ANCHORS;
         n += gridDim.x * blockDim.x) {
        unsigned s = sortable[n];
        if ((s >> 21) == sel1) atomicAdd(&lh[(s >> 10) & 0x7FFu], 1u);
    }
    __syncthreads();
    for (int t = threadIdx.x; t < 2048; t += 256)
        if (lh[t]) atomicAdd(&hist[t], lh[t]);
}

__global__ __launch_bounds__(256) void hist3_kernel(const unsigned* __restrict__ sortable,
                                                    const unsigned* __restrict__ state,
                                                    unsigned* __restrict__ hist) {
    __shared__ unsigned lh[1024];
    for (int t = threadIdx.x; t < 1024; t += 256) lh[t] = 0u;
    __syncthreads();
    const unsigned pfx = (state[ST_SEL1] << 11) | state[ST_SEL2];
    for (unsigned n = blockIdx.x * blockDim.x + threadIdx.x; n < N_ANCHORS;
         n += gridDim.x * blockDim.x) {
        unsigned s = sortable[n];
        if ((s >> 10) == pfx) atomicAdd(&lh[s & 0x3FFu], 1u);
    }
    __syncthreads();
    for (int t = threadIdx.x; t < 1024; t += 256)
        if (lh[t]) atomicAdd(&hist[t], lh[t]);
}

// ---------------- K2 scan: suffix-sum + select bin (single block) -----------
__global__ __launch_bounds__(1024) void scan_kernel(const unsigned* __restrict__ hist,
                                                    int nbins,
                                                    unsigned* __restrict__ state,
                                                    int phase) {
    __shared__ unsigned A[2048], B[2048];
    const int tid = threadIdx.x;
    for (int t = tid; t < nbins; t += 1024) A[t] = hist[t];
    __syncthreads();
    unsigned* src = A;
    unsigned* dst = B;
    for (int off = 1; off < nbins; off <<= 1) {   // inclusive suffix sums
        for (int t = tid; t < nbins; t += 1024) {
            unsigned v = src[t];
            if (t + off < nbins) v += src[t + off];
            dst[t] = v;
        }
        __syncthreads();
        unsigned* tmp = src; src = dst; dst = tmp;
    }
    const unsigned prev = (phase == 0) ? 0u : state[(phase == 1) ? ST_CA1 : ST_CA2];
    const unsigned R    = (unsigned)PRE_NMS - prev;     // remaining rank, >= 1
    for (int t = tid; t < nbins; t += 1024) {
        unsigned Sb  = src[t];
        unsigned Sb1 = (t + 1 < nbins) ? src[t + 1] : 0u;
        if (Sb >= R && Sb1 < R) {                       // unique bin
            if (phase == 0)      { state[ST_SEL1] = (unsigned)t; state[ST_CA1] = prev + Sb1; }
            else if (phase == 1) { state[ST_SEL2] = (unsigned)t; state[ST_CA2] = prev + Sb1; }
            else  state[ST_T] = (state[ST_SEL1] << 21) | (state[ST_SEL2] << 10) | (unsigned)t;
        }
    }
}

// ---------------- K3: compact keys >= T into (~score,index) pairs -----------
__global__ __launch_bounds__(256) void compact_kernel(const unsigned* __restrict__ sortable,
                                                      unsigned* __restrict__ state,
                                                      u64* __restrict__ pairs) {
    const unsigned T = state[ST_T];
    for (unsigned n = blockIdx.x * blockDim.x + threadIdx.x; n < N_ANCHORS;
         n += gridDim.x * blockDim.x) {
        unsigned s = sortable[n];
        if (s >= T) {
            unsigned slot = atomicAdd(&state[ST_CNT], 1u);
            if (slot < CAP)
                pairs[slot] = ((u64)(0xFFFFFFFFu ^ s) << 32) | (u64)n;
        }
    }
}

// ---------------- K4: single-WG 128KB-LDS bitonic sort + box re-decode ------
__global__ __launch_bounds__(1024) void sortgather_kernel(const u64* __restrict__ pairs,
                                                          const float* __restrict__ delta,
                                                          float4* __restrict__ topbox,
                                                          int* __restrict__ validf) {
    extern __shared__ u64 tile[];                 // 16384 * 8B = 128KB dynamic LDS
    const unsigned tid = threadIdx.x;
    for (unsigned t = tid; t < CAP; t += 1024) tile[t] = pairs[t];
    __syncthreads();
    for (unsigned k = 2; k <= CAP; k <<= 1) {     // ascending == desc score, asc idx
        for (unsigned j = k >> 1; j > 0; j >>= 1) {
            for (unsigned p = tid; p < CAP / 2; p += 1024) {
                unsigned i   = ((p & ~(j - 1u)) << 1) | (p & (j - 1u));
                unsigned ixj = i | j;
                bool up = ((i & k) == 0u);
                u64 a = tile[i], b = tile[ixj];
                if ((a > b) == up) { tile[i] = b; tile[ixj] = a; }
            }
            __syncthreads();
        }
    }
    // Re-decode only the 12000 winning boxes (delta is L2-resident: 192MB L2).
    for (unsigned t = tid; t < PRE_NMS; t += 1024) {
        u64 u = tile[t];
        unsigned idx = (unsigned)(u & 0xFFFFFFFFu);
        unsigned key = 0xFFFFFFFFu ^ (unsigned)(u >> 32);
        topbox[t] = decode_box(delta, idx);
        validf[t] = (key != 0u) ? 1 : 0;          // key==0 <=> score forced to -inf
    }
}

// ---------------- K5: greedy NMS, single workgroup, LDS-resident ------------
__global__ __launch_bounds__(1024) void nms_kernel(const float4* __restrict__ topbox,
                                                   const int* __restrict__ validf,
                                                   float* __restrict__ out) {
    extern __shared__ float4 sbox[];              // 12000*16B boxes + 12000*4B areas
    float* sarea = (float*)(sbox + PRE_NMS);
    __shared__ unsigned salive[PRE_NMS / 32];     // 375 words
    const int tid = threadIdx.x;

    // Fill 192KB LDS box tile via CDNA5 async global->LDS DMA (ASYNCcnt path).
    for (int t = tid; t < PRE_NMS; t += 1024) {
        unsigned ldsa = (unsigned)(uintptr_t)&sbox[t];   // low 32b = LDS byte addr
        u64 gaddr     = (u64)(uintptr_t)&topbox[t];
        asm volatile("global_load_async_to_lds_b128 %0, %1, off"
                     :: "v"(ldsa), "v"(gaddr) : "memory");
    }
    for (int wrd = tid; wrd < PRE_NMS / 32; wrd += 1024) {
        unsigned m = 0u;
        for (int b = 0; b < 32; ++b)
            if (validf[wrd * 32 + b]) m |= (1u << b);
        salive[wrd] = m;
    }
    asm volatile("s_wait_asynccnt 0" ::: "memory");
    __syncthreads();

    for (int t = tid; t < PRE_NMS; t += 1024) {   // cache areas in LDS
        float4 c = sbox[t];
        sarea[t] = (c.z - c.x) * (c.w - c.y);
    }
    __syncthreads();

    int outc = 0;                                 // uniform across all lanes
    for (int i = 0; i < PRE_NMS && outc < POST_NMS; ++i) {
        unsigned word = salive[i >> 5];
        if (!((word >> (i & 31)) & 1u)) continue; // suppressed/invalid: skip

        float4 b    = sbox[i];
        float areaB = sarea[i];
        for (int j = i + 1 + tid; j < PRE_NMS; j += 1024) {
            float4 c  = sbox[j];
            float xx0 = fmaxf(b.x, c.x), yy0 = fmaxf(b.y, c.y);
            float xx1 = fminf(b.z, c.z), yy1 = fminf(b.w, c.w);
            float inter = fmaxf(xx1 - xx0, 0.0f) * fmaxf(yy1 - yy0, 0.0f);
            float iou   = inter / fmaxf(areaB + sarea[j] - inter, 1e-8f);
            if (iou > NMS_TH)
                atomicAnd(&salive[j >> 5], ~(1u << (j & 31)));
        }
        if (tid == 0) ((float4*)out)[outc] = b;
        outc++;
        __syncthreads();    // make atomicAnds visible before next alive read
    }
    for (int t = outc + tid; t < POST_NMS; t += 1024)     // zero-fill tail
        ((float4*)out)[t] = make_float4(0.f, 0.f, 0.f, 0.f);
}

// ---------------------------------------------------------------------------
extern "C" void kernel_launch(void* const* d_in, const int* in_sizes, int n_in,
                              void* d_out, int out_size, void* d_ws, size_t ws_size,
                              hipStream_t stream) {
    const float* delta = (const float*)d_in[0];   // (1, N, 4) f32
    const float* score = (const float*)d_in[1];   // (1, N, 2) f32
    float* out = (float*)d_out;                   // 2000*4 f32

    // Workspace layout (bytes): ~2.8 MB total
    char* ws = (char*)d_ws;
    size_t off = 0;
    unsigned* sortable = (unsigned*)(ws + off); off += (size_t)N_ANCHORS * 4;   // 2.36 MB
    u64*      pairs    = (u64*)     (ws + off); off += (size_t)CAP * 8;         // 128 KB
    float4*   topbox   = (float4*)  (ws + off); off += (size_t)PRE_NMS * 16;    // 192 KB
    int*      validf   = (int*)     (ws + off); off += (size_t)PRE_NMS * 4;     //  48 KB
    unsigned* hist1    = (unsigned*)(ws + off); off += 2048 * 4;
    unsigned* hist2    = (unsigned*)(ws + off); off += 2048 * 4;
    unsigned* hist3    = (unsigned*)(ws + off); off += 1024 * 4;
    unsigned* state    = (unsigned*)(ws + off); off += 8 * 4;
    const size_t zero_bytes = (2048 + 2048 + 1024 + 8) * 4;

    // Init: zero histograms/state/counter; fill pairs pad with 0xFF (sorts last)
    (void)hipMemsetAsync(hist1, 0, zero_bytes, stream);
    (void)hipMemsetAsync(pairs, 0xFF, (size_t)CAP * 8, stream);

    // K1: sortable keys only (boxes not materialized)
    key_kernel<<<(N_ANCHORS + 255) / 256, 256, 0, stream>>>(delta, score, sortable);

    // K2: 3-level radix select -> exact threshold T in state[ST_T]
    hist1_kernel<<<256, 256, 0, stream>>>(sortable, hist1);
    scan_kernel<<<1, 1024, 0, stream>>>(hist1, 2048, state, 0);
    hist2_kernel<<<256, 256, 0, stream>>>(sortable, state, hist2);
    scan_kernel<<<1, 1024, 0, stream>>>(hist2, 2048, state, 1);
    hist3_kernel<<<256, 256, 0, stream>>>(sortable, state, hist3);
    scan_kernel<<<1, 1024, 0, stream>>>(hist3, 1024, state, 2);

    // K3: compact survivors
    compact_kernel<<<256, 256, 0, stream>>>(sortable, state, pairs);

    // K4: single-WG 128KB-LDS bitonic sort of 16384 pairs + box re-decode
    (void)hipFuncSetAttribute((const void*)sortgather_kernel,
                              hipFuncAttributeMaxDynamicSharedMemorySize, 160 * 1024);
    sortgather_kernel<<<1, 1024, CAP * sizeof(u64), stream>>>(pairs, delta, topbox, validf);

    // K5: single-WG greedy NMS, 240KB dynamic LDS (CDNA5: 320KB/WGP)
    (void)hipFuncSetAttribute((const void*)nms_kernel,
                              hipFuncAttributeMaxDynamicSharedMemorySize, 256 * 1024);
    nms_kernel<<<1, 1024, PRE_NMS * (sizeof(float4) + sizeof(float)), stream>>>(
        topbox, validf, out);
}